// TemporalFlowCell_7902739825265
// MI455X (gfx1250) — compile-verified
//
#include <hip/hip_runtime.h>
#include <math.h>

typedef float v2f __attribute__((ext_vector_type(2)));
typedef float v8f __attribute__((ext_vector_type(8)));

#define WMMA_F32(a, b, c) \
  __builtin_amdgcn_wmma_f32_16x16x4_f32(false, (a), false, (b), (short)0, (c), false, false)

static constexpr int B_ = 8, S_ = 2048, D_ = 1024, E_ = 64;
static constexpr int M_ = B_ * S_;           // 16384 rows
static constexpr int NCHUNK = 32, CH = 64;
static constexpr long long YSZ = (long long)M_ * D_;

// ---------------------------------------------------------------------------
// Kernel 1: beta = x @ w_in^T   (16384x1024)(1024x64), f32 WMMA 16x16x4
// One wave = one 16-row M tile over all N=64 (4 n-tiles). 4 waves / block.
// ---------------------------------------------------------------------------
__global__ __launch_bounds__(128)
void k_beta_gemm(const float* __restrict__ x, const float* __restrict__ w_in,
                 float* __restrict__ beta) {
  const int lane = threadIdx.x & 31;
  const int wv   = (blockIdx.x << 2) + (threadIdx.x >> 5); // global wave id: 0..1023
  const int m16  = lane & 15;
  const int half = lane >> 4;

  const float* xr = x + (size_t)(wv * 16 + m16) * D_ + 2 * half;

  v8f acc[4];
#pragma unroll
  for (int i = 0; i < 4; ++i) acc[i] = {};

  for (int kb = 0; kb < D_; kb += 4) {
    float2 av = *(const float2*)(xr + kb);      // A[m][kb+2h], A[m][kb+2h+1]
    v2f a; a.x = av.x; a.y = av.y;
#pragma unroll
    for (int nt = 0; nt < 4; ++nt) {
      const int e = nt * 16 + m16;              // B n-index = e
      float2 bv = *(const float2*)(w_in + (size_t)e * D_ + kb + 2 * half);
      v2f bb; bb.x = bv.x; bb.y = bv.y;
      acc[nt] = WMMA_F32(a, bb, acc[nt]);
    }
  }
  // C/D layout: lane n = m16, VGPR v -> row v + 8*half
#pragma unroll
  for (int nt = 0; nt < 4; ++nt)
#pragma unroll
    for (int v = 0; v < 8; ++v)
      beta[(size_t)(wv * 16 + v + 8 * half) * E_ + nt * 16 + m16] = acc[nt][v];
}

// ---------------------------------------------------------------------------
// Kernels 2/3: chunk-local scans. One block per chunk (512 thr = (b,e) pairs).
// phase 0: zero-init, emit chunk-final complex state F[c].
// phase 1: seed with F[c-1] (||A^64|| <= ~0.5^64 < 1e-19 => deeper carry is
//          exactly 0 in f32), write all states + final r/im for chunk 31.
// Inner product uses packed (re,im) math -> V_PK_FMA_F32, and the beta load
// for step tau+1 is issued before step tau's barrier+accumulation so global
// latency never lands on the serial path.
// ---------------------------------------------------------------------------
__global__ __launch_bounds__(512)
void k_scan(const float* __restrict__ beta, const float* __restrict__ alpha,
            const float* __restrict__ omega, const float* __restrict__ res,
            float2* __restrict__ F, float* __restrict__ states,
            float* __restrict__ rfin, float* __restrict__ ifin, int phase) {
  const int c = blockIdx.x;
  const int t = threadIdx.x;
  const int b = t >> 6;
  const int e = t & 63;

  const float mag = 1.f / (1.f + expf(-alpha[e]));
  const float cw  = cosf(omega[e]);
  const float sw  = sinf(omega[e]);

  // resonance column e held in registers: rc[k] = R[k][e]
  float rc[64];
#pragma unroll
  for (int k = 0; k < 64; ++k) rc[k] = res[k * E_ + e];

  __shared__ float buf[2][B_ * E_ * 2];   // (nr,ni) interleaved, double-buffered

  float re = 0.f, im = 0.f;
  if (phase && c > 0) {
    float2 z = F[((c - 1) * B_ + b) * E_ + e];
    re = z.x; im = z.y;
  }

  const float* bp = beta   + (size_t)(b * S_ + c * CH) * E_ + e;
  float*       sp = states + (size_t)(b * S_ + c * CH) * E_ + e;

  float bt = bp[0];                        // software pipeline: beta one step ahead
  for (int tau = 0; tau < CH; ++tau) {
    const int nxt = (tau + 1 < CH) ? (tau + 1) : (CH - 1);
    const float bt_next = bp[nxt * E_];    // issued early, consumed next iter

    const float nr = fmaf(mag, re * cw - im * sw, bt);
    const float ni = mag * (re * sw + im * cw);
    *(float2*)&buf[tau & 1][(b * E_ + e) * 2] = make_float2(nr, ni);
    __syncthreads();

    v2f acc0 = {0.f, 0.f}, acc1 = {0.f, 0.f};
    const float4* rd = (const float4*)&buf[tau & 1][b * E_ * 2];
#pragma unroll
    for (int k = 0; k < 64; k += 4) {
      float4 va = rd[k >> 1];              // broadcast reads: (nr,ni) pairs k..k+1
      float4 vb = rd[(k >> 1) + 1];        // pairs k+2..k+3
      v2f p0 = {va.x, va.y}, p1 = {va.z, va.w};
      v2f p2 = {vb.x, vb.y}, p3 = {vb.z, vb.w};
      v2f r0 = {rc[k], rc[k]},         r1 = {rc[k + 1], rc[k + 1]};
      v2f r2 = {rc[k + 2], rc[k + 2]}, r3 = {rc[k + 3], rc[k + 3]};
      acc0 = __builtin_elementwise_fma(p0, r0, acc0);   // -> v_pk_fma_f32
      acc1 = __builtin_elementwise_fma(p1, r1, acc1);
      acc0 = __builtin_elementwise_fma(p2, r2, acc0);
      acc1 = __builtin_elementwise_fma(p3, r3, acc1);
    }
    const v2f accs = acc0 + acc1;
    re = accs.x; im = accs.y;
    if (phase) sp[tau * E_] = re;          // reference stores post-resonance r
    bt = bt_next;
  }

  if (!phase) {
    F[(c * B_ + b) * E_ + e] = make_float2(re, im);
  } else if (c == NCHUNK - 1) {
    rfin[b * E_ + e] = re;
    ifin[b * E_ + e] = im;
  }
}

// ---------------------------------------------------------------------------
// Kernel 4: y = LN(states @ w_out^T). One block = 16 rows x full D=1024,
// fused through LDS so y_pre never touches HBM. f32 WMMA, K=64 (16 k-steps).
// ---------------------------------------------------------------------------
__global__ __launch_bounds__(128)
void k_out_ln(const float* __restrict__ states, const float* __restrict__ w_out,
              const float* __restrict__ gamma, const float* __restrict__ lbeta,
              float* __restrict__ y) {
  const int lane = threadIdx.x & 31;
  const int wv   = threadIdx.x >> 5;        // 0..3
  const int m16  = lane & 15;
  const int half = lane >> 4;
  const int rowbase = blockIdx.x * 16;

  __shared__ float  yt[16][D_ + 4];
  __shared__ float  psum[16][8], psq[16][8];
  __shared__ float2 lnp[16];

  // A (16 rows x K=64 of states), all 16 k-steps hoisted into registers
  v2f a[16];
  const float* srow = states + (size_t)(rowbase + m16) * E_ + 2 * half;
#pragma unroll
  for (int ks = 0; ks < 16; ++ks) {
    float2 av = *(const float2*)(srow + 4 * ks);
    a[ks].x = av.x; a[ks].y = av.y;
  }

  for (int dt = wv * 16; dt < wv * 16 + 16; ++dt) {   // 16 d-tiles per wave
    v8f acc = {};
#pragma unroll
    for (int ks = 0; ks < 16; ++ks) {
      // B[k][n] = w_out[d][e], d = dt*16+m16, e = 4*ks + 2*half (+1)
      float2 bv = *(const float2*)(w_out + (size_t)(dt * 16 + m16) * E_ + 4 * ks + 2 * half);
      v2f bb; bb.x = bv.x; bb.y = bv.y;
      acc = WMMA_F32(a[ks], bb, acc);
    }
#pragma unroll
    for (int v = 0; v < 8; ++v)
      yt[v + 8 * half][dt * 16 + m16] = acc[v];
  }
  __syncthreads();

  // LayerNorm over D per row: 8 threads/row partial sums
  {
    const int r = threadIdx.x >> 3, ss = threadIdx.x & 7;
    float s0 = 0.f, s1 = 0.f;
    for (int j = ss; j < D_; j += 8) {
      float v = yt[r][j];
      s0 += v; s1 += v * v;
    }
    psum[r][ss] = s0; psq[r][ss] = s1;
  }
  __syncthreads();
  if (threadIdx.x < 16) {
    const int r = threadIdx.x;
    float s0 = 0.f, s1 = 0.f;
#pragma unroll
    for (int j = 0; j < 8; ++j) { s0 += psum[r][j]; s1 += psq[r][j]; }
    const float mu  = s0 * (1.f / D_);
    const float var = s1 * (1.f / D_) - mu * mu;
    lnp[r] = make_float2(mu, rsqrtf(var + 1e-5f));
  }
  __syncthreads();

  for (int r = 0; r < 16; ++r) {
    const float mu = lnp[r].x, rs = lnp[r].y;
    for (int j = threadIdx.x; j < D_; j += 128) {
      float v = (yt[r][j] - mu) * rs * gamma[j] + lbeta[j];
      y[(size_t)(rowbase + r) * D_ + j] = v;
    }
  }
}

// ---------------------------------------------------------------------------
extern "C" void kernel_launch(void* const* d_in, const int* in_sizes, int n_in,
                              void* d_out, int out_size, void* d_ws, size_t ws_size,
                              hipStream_t stream) {
  (void)in_sizes; (void)n_in; (void)out_size; (void)ws_size;

  const float* x     = (const float*)d_in[0];
  const float* alpha = (const float*)d_in[1];
  const float* omega = (const float*)d_in[2];
  const float* w_in  = (const float*)d_in[3];
  const float* w_out = (const float*)d_in[4];
  const float* res   = (const float*)d_in[5];
  const float* gamma = (const float*)d_in[6];
  const float* lbeta = (const float*)d_in[7];

  float* out  = (float*)d_out;
  float* y    = out;
  float* rfin = out + YSZ;
  float* ifin = rfin + B_ * E_;

  float*  beta   = (float*)d_ws;                        // 16384*64 f32 = 4 MB
  float*  states = beta + (size_t)M_ * E_;              // 4 MB
  float2* F      = (float2*)(states + (size_t)M_ * E_); // 32*8*64 float2 = 128 KB

  // 1) beta = x @ w_in^T
  hipLaunchKernelGGL(k_beta_gemm, dim3(M_ / 64), dim3(128), 0, stream, x, w_in, beta);
  // 2) parallel zero-init chunk scans -> chunk finals F
  hipLaunchKernelGGL(k_scan, dim3(NCHUNK), dim3(512), 0, stream,
                     beta, alpha, omega, res, F, states, rfin, ifin, 0);
  // 3) seeded chunk scans -> exact states + final (r, im)
  hipLaunchKernelGGL(k_scan, dim3(NCHUNK), dim3(512), 0, stream,
                     beta, alpha, omega, res, F, states, rfin, ifin, 1);
  // 4) y = LayerNorm(states @ w_out^T), fused
  hipLaunchKernelGGL(k_out_ln, dim3(M_ / 16), dim3(128), 0, stream,
                     states, w_out, gamma, lbeta, y);
}